// TopKRouter_77756087927339
// MI455X (gfx1250) — compile-verified
//
#include <hip/hip_runtime.h>
#include <hip/hip_bf16.h>

// ---------------- constants ----------------
#define D_DIM     2048
#define E_DIM     64
#define TOKENS    16384          // B*S = 4*4096
#define TOP_K     2
#define AUX_COEF  0.01f
#define KC        64             // D-chunk staged in LDS
#define NCHUNK    (D_DIM / KC)   // 32
#define PITCH     68             // padded LDS row pitch (floats); 272B rows, 16B aligned
#define TOK_PER_BLOCK 128        // 8 waves * 16 tokens
#define GOFF      (TOKENS * TOP_K)   // gate offset in d_out
#define AUXOFF    (2 * TOKENS * TOP_K)

typedef __attribute__((ext_vector_type(2))) float v2f;
typedef __attribute__((ext_vector_type(8))) float v8f;

// ---- CDNA5 async global->LDS copy (ASYNCcnt-tracked, no VGPR round-trip) ----
// VGLOBAL encoding: VDST = VGPR holding LDS byte address, VADDR = 64-bit global
// address pair, SADDR = off (GV mode). Low 32 bits of a generic shared-memory
// pointer are the LDS byte offset (flat LDS aperture maps addr[31:0] -> LDS).
static __device__ __forceinline__ void async_copy_b128(void* lds_dst, const void* gsrc) {
    unsigned int lds_off = (unsigned int)(unsigned long long)lds_dst;
    asm volatile("global_load_async_to_lds_b128 %0, %1, off"
                 :: "v"(lds_off), "v"(gsrc)
                 : "memory");
}
#define ASYNC_WAIT_12() asm volatile("s_wait_asynccnt 12" ::: "memory")
#define ASYNC_WAIT_0()  asm volatile("s_wait_asynccnt 0"  ::: "memory")

// top-2 insert with lax.top_k tie-break (lower index wins on equal value)
static __device__ __forceinline__ void top2_insert(float v, int i,
                                                   float& v1, int& i1,
                                                   float& v2, int& i2) {
    if (v > v1 || (v == v1 && i < i1)) {
        v2 = v1; i2 = i1;
        v1 = v;  i1 = i;
    } else if (v > v2 || (v == v2 && i < i2)) {
        v2 = v;  i2 = i;
    }
}

__global__ void __launch_bounds__(256)
router_zero_ws(float* __restrict__ ws) {
    // ws[0..63] = counts per expert, ws[64..127] = sum of router probs per expert
    int t = threadIdx.x;
    if (t < 2 * E_DIM) ws[t] = 0.0f;
}

// stage one D-chunk (x: 128x64, W: 64x64) into LDS via async b128 copies.
// 12 async instructions per thread => per-wave ASYNCcnt += 12.
static __device__ __forceinline__ void stage_chunk_async(
        const float* __restrict__ x, const float* __restrict__ W,
        float* __restrict__ xsbuf, float* __restrict__ wsbuf,
        int tokBase, int d0, int tid) {
    #pragma unroll
    for (int i = 0; i < 8; ++i) {
        int idx = tid + i * 256;            // 0..2047 float4 slots
        int r   = idx >> 4;                 // token row 0..127
        int c   = (idx & 15) << 2;          // col 0..60 step 4
        async_copy_b128(xsbuf + r * PITCH + c,
                        x + (size_t)(tokBase + r) * D_DIM + d0 + c);
    }
    #pragma unroll
    for (int i = 0; i < 4; ++i) {
        int idx = tid + i * 256;            // 0..1023 float4 slots
        int r   = idx >> 4;                 // expert row 0..63
        int c   = (idx & 15) << 2;
        async_copy_b128(wsbuf + r * PITCH + c,
                        W + (size_t)r * D_DIM + d0 + c);
    }
}

__global__ void __launch_bounds__(256)
router_main(const float* __restrict__ x,   // [TOKENS, D]
            const float* __restrict__ W,   // [E, D]
            float* __restrict__ out,       // [2*TOKENS*K + 1]
            float* __restrict__ aux_ws) {  // [128] floats
    __shared__ float xs[2][TOK_PER_BLOCK * PITCH]; // 2 x (128 x 64+pad) fp32
    __shared__ float wsh[2][E_DIM * PITCH];        // 2 x ( 64 x 64+pad) fp32

    const int tid    = threadIdx.x;
    const int lane   = tid & 31;
    const int wave   = tid >> 5;          // 0..7
    const int laneLo = lane & 15;         // N / M index within tile
    const int hi     = lane >> 4;         // K-half selector for A/B fragments
    const int tokBase = blockIdx.x * TOK_PER_BLOCK;

    v8f acc[4] = {};                      // 16x64 logits per wave (4 N-tiles)
    float psum[4] = {0.f, 0.f, 0.f, 0.f}; // per-lane expert prob sums (aux loss)

    // ---- prologue: prefetch chunks 0 and 1 (double buffered) ----
    stage_chunk_async(x, W, xs[0], wsh[0], tokBase, 0,  tid);
    stage_chunk_async(x, W, xs[1], wsh[1], tokBase, KC, tid);
    ASYNC_WAIT_12();      // chunk 0 landed (async loads complete in order)
    __syncthreads();

    for (int c = 0; c < NCHUNK; ++c) {
        const float* __restrict__ xb = xs[c & 1];
        const float* __restrict__ wb = wsh[c & 1];

        // ---- WMMA accumulate: A = x[16 tok x 4k], B = W^T[4k x 16 exp] ----
        const float* xrow = &xb[(wave * 16 + laneLo) * PITCH];
        #pragma unroll
        for (int k0 = 0; k0 < KC; k0 += 4) {
            const int ka = k0 + 2 * hi;     // lanes 0-15: K0/K1, 16-31: K2/K3
            v2f a = *(const v2f*)(xrow + ka);
            #pragma unroll
            for (int e = 0; e < 4; ++e) {
                v2f b = *(const v2f*)(&wb[(e * 16 + laneLo) * PITCH + ka]);
                acc[e] = __builtin_amdgcn_wmma_f32_16x16x4_f32(
                    /*neg_a=*/false, a, /*neg_b=*/false, b,
                    /*c_mod=*/(short)0, acc[e],
                    /*reuse_a=*/false, /*reuse_b=*/false);
            }
        }

        __syncthreads();  // all waves done reading buf[c&1]
        if (c + 2 < NCHUNK) {
            // refill the buffer we just finished with chunk c+2, then make
            // sure chunk c+1 (issued one iteration earlier) has landed.
            stage_chunk_async(x, W, xs[c & 1], wsh[c & 1], tokBase,
                              (c + 2) * KC, tid);
            ASYNC_WAIT_12();
            __syncthreads();
        } else if (c + 1 < NCHUNK) {
            ASYNC_WAIT_0();
            __syncthreads();
        }
    }

    // ---- softmax + top-2 + gates + aux stats over E=64 per token row ----
    // C layout: VGPR j holds M=j (lanes 0-15) and M=j+8 (lanes 16-31); N = t*16 + laneLo.
    #pragma unroll
    for (int j = 0; j < 8; ++j) {
        float v[4];
        #pragma unroll
        for (int t = 0; t < 4; ++t) v[t] = acc[t][j];

        float v1 = -__builtin_huge_valf(), v2 = -__builtin_huge_valf();
        int   i1 = 0x7fffffff, i2 = 0x7fffffff;
        #pragma unroll
        for (int t = 0; t < 4; ++t) top2_insert(v[t], t * 16 + laneLo, v1, i1, v2, i2);

        // butterfly over the 16-lane half (masks <= 8 stay within the half)
        #pragma unroll
        for (int m = 1; m <= 8; m <<= 1) {
            float ov1 = __shfl_xor(v1, m, 32);
            int   oi1 = __shfl_xor(i1, m, 32);
            float ov2 = __shfl_xor(v2, m, 32);
            int   oi2 = __shfl_xor(i2, m, 32);
            top2_insert(ov1, oi1, v1, i1, v2, i2);
            top2_insert(ov2, oi2, v1, i1, v2, i2);
        }

        const float mx = v1;                 // global max over 64 experts
        float ex[4];
        float s = 0.f;
        #pragma unroll
        for (int t = 0; t < 4; ++t) { ex[t] = __expf(v[t] - mx); s += ex[t]; }
        #pragma unroll
        for (int m = 1; m <= 8; m <<= 1) s += __shfl_xor(s, m, 32);
        const float inv = 1.0f / s;

        #pragma unroll
        for (int t = 0; t < 4; ++t) psum[t] += ex[t] * inv; // full softmax probs (aux)

        if (laneLo == j) {  // lane j -> token j (half 0), lane 16+j -> token j+8
            const float p1 = inv;                    // exp(0)/s
            const float p2 = __expf(v2 - mx) * inv;
            const float dn = p1 + p2 + 1e-9f;
            const int tok = tokBase + wave * 16 + j + 8 * hi;
            out[2 * tok]            = (float)i1;
            out[2 * tok + 1]        = (float)i2;
            out[GOFF + 2 * tok]     = p1 / dn;
            out[GOFF + 2 * tok + 1] = p2 / dn;
            atomicAdd(&aux_ws[i1], 1.0f);            // routed-slot counts
            atomicAdd(&aux_ws[i2], 1.0f);
        }
    }

    // accumulate router-prob sums per expert (both halves hit the same expert id)
    #pragma unroll
    for (int t = 0; t < 4; ++t)
        atomicAdd(&aux_ws[E_DIM + t * 16 + laneLo], psum[t]);
}

__global__ void __launch_bounds__(64)
router_aux_finalize(const float* __restrict__ aux_ws, float* __restrict__ out) {
    if (threadIdx.x == 0) {
        float s = 0.f;
        const float inv_slots  = 1.0f / (float)(TOKENS * TOP_K);
        const float inv_tokens = 1.0f / (float)TOKENS;
        for (int e = 0; e < E_DIM; ++e)
            s += (aux_ws[e] * inv_slots) * (aux_ws[E_DIM + e] * inv_tokens);
        out[AUXOFF] = AUX_COEF * (float)E_DIM * s;
    }
}

extern "C" void kernel_launch(void* const* d_in, const int* in_sizes, int n_in,
                              void* d_out, int out_size, void* d_ws, size_t ws_size,
                              hipStream_t stream) {
    (void)in_sizes; (void)n_in; (void)out_size; (void)ws_size;
    const float* x = (const float*)d_in[0];   // [4,4096,2048] fp32
    const float* W = (const float*)d_in[1];   // [64,2048] fp32
    float* out = (float*)d_out;
    float* ws  = (float*)d_ws;                // 128 floats: counts[64] + probsum[64]

    router_zero_ws<<<1, 256, 0, stream>>>(ws);
    router_main<<<TOKENS / TOK_PER_BLOCK, 256, 0, stream>>>(x, W, out, ws);
    router_aux_finalize<<<1, 64, 0, stream>>>(ws, out);
}